// GIN_87265145520538
// MI455X (gfx1250) — compile-verified
//
#include <hip/hip_runtime.h>

typedef float v2f __attribute__((ext_vector_type(2)));
typedef float v8f __attribute__((ext_vector_type(8)));

#define D 128
#define NGRAPH 128

// ---------------- elementwise helpers ----------------

__global__ __launch_bounds__(256) void copy_f4(float* __restrict__ dst,
                                               const float* __restrict__ src,
                                               long long n4) {
  long long t = (long long)blockIdx.x * 256 + threadIdx.x;
  if (t < n4) ((float4*)dst)[t] = ((const float4*)src)[t];
}

__global__ __launch_bounds__(256) void zero_f(float* __restrict__ p, int n) {
  int t = blockIdx.x * 256 + threadIdx.x;
  if (t < n) p[t] = 0.f;
}

// ---------------- edge scatter-add: h[dst] += z[src] ----------------
// 32 lanes per edge, float4 per lane (128 cols). Edge indices are wave-uniform
// (all 32 lanes of a wave share one edge) -> scalar loads.

__global__ __launch_bounds__(256) void scatter_add(float* __restrict__ h,
                                                   const float* __restrict__ z,
                                                   const int* __restrict__ ei,
                                                   int n_edges) {
  long long t = (long long)blockIdx.x * 256 + threadIdx.x;
  int e = (int)(t >> 5);
  if (e >= n_edges) return;
  int lane = (int)(t & 31);
  int s = ei[e];            // src row
  int d = ei[n_edges + e];  // dst row
  const float4 v = *(const float4*)(z + (size_t)s * D + lane * 4);
  float* p = h + (size_t)d * D + lane * 4;
  atomicAdd(p + 0, v.x);
  atomicAdd(p + 1, v.y);
  atomicAdd(p + 2, v.z);
  atomicAdd(p + 3, v.w);
}

// ---------------- fused GEMM + bias + ReLU via V_WMMA_F32_16X16X4_F32 ----------
// out[M x 128] = relu(in[M x 128] @ W[128 x 128] + bias[128])
// Block: 256 threads = 8 waves. Wave w owns output cols [16w, 16w+16).
// Block owns 64 output rows (4 sequential 16-row tiles per wave).
// W staged in LDS (64 KB).
//
// f32 WMMA fragment layouts (ISA 7.12.2):
//   A 16x4 : lane half h, VGPR v -> (M = lane&15, K = 2h + v)
//   B 4x16 : lane half h, VGPR v -> (K = 2h + v, N = lane&15)
//   C 16x16: lane half h, VGPR v -> (M = v + 8h, N = lane&15)

__global__ __launch_bounds__(256) void gemm_bias_relu(float* __restrict__ out,
                                                      const float* __restrict__ in,
                                                      const float* __restrict__ W,
                                                      const float* __restrict__ bias,
                                                      int M) {
  __shared__ float Wsh[D * D];
  for (int i = threadIdx.x; i < D * D / 4; i += 256)
    ((float4*)Wsh)[i] = ((const float4*)W)[i];
  __syncthreads();

  const int wave = threadIdx.x >> 5;
  const int lane = threadIdx.x & 31;
  const int half = lane >> 4;
  const int lm = lane & 15;
  const int n0 = wave * 16;
  const float bv = bias[n0 + lm];
  const int mbase = blockIdx.x * 64;

  for (int mt = 0; mt < 4; ++mt) {
    const int m0 = mbase + mt * 16;
    if (m0 + 16 > M) break;  // M is a multiple of 16
    v8f c = {};
    const float* arow = in + (size_t)(m0 + lm) * D + 2 * half;
#pragma unroll
    for (int k0 = 0; k0 < D; k0 += 4) {
      v2f a = *(const v2f*)(arow + k0);  // A[M=m0+lm][k0+2h .. +1]
      v2f b;
      b.x = Wsh[(k0 + 2 * half) * D + n0 + lm];      // B[k0+2h][n]
      b.y = Wsh[(k0 + 2 * half + 1) * D + n0 + lm];  // B[k0+2h+1][n]
      c = __builtin_amdgcn_wmma_f32_16x16x4_f32(false, a, false, b,
                                                (short)0, c, false, false);
    }
#pragma unroll
    for (int v = 0; v < 8; ++v) {
      const int m = m0 + v + 8 * half;
      float r = c[v] + bv;
      out[(size_t)m * D + n0 + lm] = r > 0.f ? r : 0.f;
    }
  }
}

// ---------------- segment-mean pool (batch is sorted) ----------------
// 128 threads (one per column) per block; each block scans 256 contiguous
// nodes, keeping a register running sum per graph-run; flushes via atomics
// only on graph-id change (~2 flushes/block instead of 256).

__global__ __launch_bounds__(128) void pool_sum(const float* __restrict__ z,
                                                const int* __restrict__ batch,
                                                float* __restrict__ sums,
                                                float* __restrict__ counts,
                                                int n) {
  const int c = threadIdx.x;  // column 0..127
  int start = blockIdx.x * 256;
  if (start >= n) return;
  int end = start + 256;
  if (end > n) end = n;
  int g = batch[start];
  float acc = 0.f, run = 0.f;
  for (int i = start; i < end; ++i) {
    int gi = batch[i];
    if (gi != g) {
      atomicAdd(sums + (size_t)g * D + c, acc);
      if (c == 0) atomicAdd(counts + g, run);
      acc = 0.f;
      run = 0.f;
      g = gi;
    }
    acc += z[(size_t)i * D + c];
    run += 1.f;
  }
  atomicAdd(sums + (size_t)g * D + c, acc);
  if (c == 0) atomicAdd(counts + g, run);
}

__global__ __launch_bounds__(256) void pool_div(float* __restrict__ gout,
                                                const float* __restrict__ sums,
                                                const float* __restrict__ counts,
                                                int total) {
  int t = blockIdx.x * 256 + threadIdx.x;
  if (t < total) {
    float cnt = counts[t >> 7];
    gout[t] = sums[t] / (cnt > 1.f ? cnt : 1.f);
  }
}

// ---------------- driver ----------------

extern "C" void kernel_launch(void* const* d_in, const int* in_sizes, int n_in,
                              void* d_out, int out_size, void* d_ws, size_t ws_size,
                              hipStream_t stream) {
  const float* x    = (const float*)d_in[0];
  const int*   ei   = (const int*)d_in[1];  // (2, E): row0=src, row1=dst
  const int*   batch= (const int*)d_in[2];
  const float* W1   = (const float*)d_in[3];
  const float* b1   = (const float*)d_in[4];
  const float* W2   = (const float*)d_in[5];
  const float* b2   = (const float*)d_in[6];

  const int n       = in_sizes[2];
  const int n_edges = in_sizes[1] / 2;
  const int L       = in_sizes[3] / (D * D);
  const size_t feat = (size_t)n * D;

  float* out_z  = (float*)d_out;
  float* out_g  = out_z + feat;
  float* bufA   = (float*)d_ws;
  float* bufB   = bufA + feat;
  float* sums   = bufB + feat;           // NGRAPH*D
  float* counts = sums + NGRAPH * D;     // NGRAPH

  const long long n4   = (long long)feat / 4;
  const int copyBlocks = (int)((n4 + 255) / 256);
  const int scatBlocks = (int)(((long long)n_edges * 32 + 255) / 256);
  const int gemmBlocks = (n + 63) / 64;
  const int poolBlocks = (n + 255) / 256;

  const float* zcur = x;
  float* hbuf = bufA;
  float* tbuf = bufB;
  for (int l = 0; l < L; ++l) {
    // h = z + sum_{src->dst} z[src]
    copy_f4<<<copyBlocks, 256, 0, stream>>>(hbuf, zcur, n4);
    scatter_add<<<scatBlocks, 256, 0, stream>>>(hbuf, zcur, ei, n_edges);
    // h2 = relu(h @ W1[l] + b1[l])
    gemm_bias_relu<<<gemmBlocks, 256, 0, stream>>>(tbuf, hbuf,
        W1 + (size_t)l * D * D, b1 + (size_t)l * D, n);
    // z = relu(h2 @ W2[l] + b2[l]); last layer writes d_out directly
    float* zdst = (l == L - 1) ? out_z : hbuf;
    gemm_bias_relu<<<gemmBlocks, 256, 0, stream>>>(zdst, tbuf,
        W2 + (size_t)l * D * D, b2 + (size_t)l * D, n);
    zcur = zdst;
    float* tmp = hbuf; hbuf = tbuf; tbuf = tmp;
  }

  // segment-mean pool
  zero_f<<<(NGRAPH * D + NGRAPH + 255) / 256, 256, 0, stream>>>(
      sums, NGRAPH * D + NGRAPH);
  pool_sum<<<poolBlocks, 128, 0, stream>>>(zcur, batch, sums, counts, n);
  pool_div<<<(NGRAPH * D + 255) / 256, 256, 0, stream>>>(out_g, sums, counts,
                                                         NGRAPH * D);
}